// KM_UNet_80247168958521
// MI455X (gfx1250) — compile-verified
//
#include <hip/hip_runtime.h>
#include <math.h>

typedef __attribute__((ext_vector_type(2))) float v2f;
typedef __attribute__((ext_vector_type(8))) float v8f;

#define EPSV 1e-5f
#define ACT_NONE 0
#define ACT_RELU 1
#define ACT_SILU 2

__device__ __forceinline__ float sigf(float x) { return 1.f / (1.f + __expf(-x)); }
__device__ __forceinline__ float actf(float v, int act) {
  if (act == ACT_RELU) return fmaxf(v, 0.f);
  if (act == ACT_SILU) return v * sigf(v);
  return v;
}
__device__ __forceinline__ float softplusf(float x) {
  return (x > 20.f) ? x : log1pf(__expf(x));
}

// ---------------------------------------------------------------------------
// WMMA fp32 GEMM:  Out = epilogue( A[M,K] * B[K,N] )
// Output indexing is NCHW-aware: column n -> (b = n/HW, hw = n%HW),
// out[((b*M + m)*HW) + hw].  Pass HW=N for a plain row-major [M][N] output.
// Tiles are staged into LDS with GLOBAL_LOAD_ASYNC_TO_LDS_B128 (ASYNCcnt)
// whenever the tile is fully in-bounds and 16B-aligned; ragged tiles use a
// guarded scalar path.
// ---------------------------------------------------------------------------
#define GT_M 32
#define GT_N 64
#define GT_K 16

__global__ __launch_bounds__(256) void k_gemm(
    const float* __restrict__ A, const float* __restrict__ B, float* __restrict__ Out,
    int M, int N, int K,
    const float* __restrict__ alpha, const float* __restrict__ beta,
    int act, int accum, int HW) {
  __shared__ float As[GT_M][GT_K + 1];
  __shared__ float Bs[GT_K][GT_N + 1];
  int tid = threadIdx.x;
  int wave = tid >> 5;
  int lane = tid & 31;
  int wm = wave >> 2;   // 0..1
  int wn = wave & 3;    // 0..3
  int bm = blockIdx.y * GT_M;
  int bn = blockIdx.x * GT_N;
  int half = lane >> 4;
  int l16 = lane & 15;

  // uniform: async fast path legal for this problem shape?
  bool seg_ok = ((K & 3) == 0) && ((N & 3) == 0);

  v8f acc = {};
  for (int k0 = 0; k0 < K; k0 += GT_K) {
    bool full = seg_ok && (k0 + GT_K <= K) && (bm + GT_M <= M) && (bn + GT_N <= N);
    if (full) {
      // ---- async DMA: global -> LDS, 16B segments ----
      if (tid < 128) {  // A tile: 32 rows x 4 segments
        int r = tid >> 2;
        int c4 = (tid & 3) * 4;
        unsigned long long ga = (unsigned long long)(A + (size_t)(bm + r) * K + k0 + c4);
        unsigned la = (unsigned)(uintptr_t)&As[r][c4];
        asm volatile("global_load_async_to_lds_b128 %0, %1, off"
                     :: "v"(la), "v"(ga) : "memory");
      }
      {  // B tile: 16 rows x 16 segments
        int r = tid >> 4;
        int c4 = (tid & 15) * 4;
        unsigned long long gb = (unsigned long long)(B + (size_t)(k0 + r) * N + bn + c4);
        unsigned lb = (unsigned)(uintptr_t)&Bs[r][c4];
        asm volatile("global_load_async_to_lds_b128 %0, %1, off"
                     :: "v"(lb), "v"(gb) : "memory");
      }
      // prefetch next K tile of B (speculative; dropped if invalid)
      if (k0 + GT_K < K) {
        __builtin_prefetch(B + (size_t)(k0 + GT_K + (tid >> 4)) * N + bn + (tid & 15) * 4, 0, 0);
      }
      asm volatile("s_wait_asynccnt 0x0" ::: "memory");
    } else {
      // ---- guarded scalar path (zero-fill) ----
      for (int i = tid; i < GT_M * GT_K; i += 256) {
        int r = i / GT_K, c = i % GT_K;
        int gr = bm + r, gc = k0 + c;
        As[r][c] = (gr < M && gc < K) ? A[(size_t)gr * K + gc] : 0.f;
      }
      for (int i = tid; i < GT_K * GT_N; i += 256) {
        int r = i / GT_N, c = i % GT_N;
        int gr = k0 + r, gc = bn + c;
        Bs[r][c] = (gr < K && gc < N) ? B[(size_t)gr * N + gc] : 0.f;
      }
    }
    __syncthreads();
#pragma unroll
    for (int kk = 0; kk < GT_K; kk += 4) {
      // A frag: lane -> (Mrow = l16, Kcol = v + 2*half); B frag: (Kcol same, Ncol = l16)
      v2f af, bf;
      af.x = As[wm * 16 + l16][kk + 0 + 2 * half];
      af.y = As[wm * 16 + l16][kk + 1 + 2 * half];
      bf.x = Bs[kk + 0 + 2 * half][wn * 16 + l16];
      bf.y = Bs[kk + 1 + 2 * half][wn * 16 + l16];
      acc = __builtin_amdgcn_wmma_f32_16x16x4_f32(false, af, false, bf,
                                                  (short)0, acc, false, false);
    }
    __syncthreads();
  }
  int n = bn + wn * 16 + l16;
  if (n < N) {
    int b = n / HW, hw = n % HW;
#pragma unroll
    for (int r = 0; r < 8; ++r) {
      int m = bm + wm * 16 + r + 8 * half;
      if (m < M) {
        size_t oi = ((size_t)b * M + m) * (size_t)HW + hw;
        float v = acc[r];
        v = v * (alpha ? alpha[m] : 1.f) + (beta ? beta[m] : 0.f);
        if (accum) v += Out[oi];
        Out[oi] = actf(v, act);
      }
    }
  }
}

// ---------------------------------------------------------------------------
// im2col for 3x3, pad=1, stride s.  col[(ci*9 + ky*3 + kx)][b*Ho*Wo + oy*Wo + ox]
// ---------------------------------------------------------------------------
__global__ void k_im2col(float* __restrict__ col, const float* __restrict__ x,
                         int B, int C, int H, int W, int Ho, int Wo, int stride) {
  size_t Ntot = (size_t)B * Ho * Wo;
  size_t idx = (size_t)blockIdx.x * blockDim.x + threadIdx.x;
  size_t tot = (size_t)C * 9 * Ntot;
  if (idx >= tot) return;
  size_t n = idx % Ntot;
  size_t kk = idx / Ntot;
  int r = (int)(kk % 9);
  int ci = (int)(kk / 9);
  int ky = r / 3, kx = r % 3;
  int b = (int)(n / ((size_t)Ho * Wo));
  int rem = (int)(n % ((size_t)Ho * Wo));
  int oy = rem / Wo, ox = rem % Wo;
  int iy = oy * stride - 1 + ky;
  int ix = ox * stride - 1 + kx;
  float v = 0.f;
  if (iy >= 0 && iy < H && ix >= 0 && ix < W)
    v = x[(((size_t)b * C + ci) * H + iy) * W + ix];
  col[idx] = v;
}

// BN fold: alpha = g/sqrt(v+eps); beta = convb*alpha + b - m*alpha
__global__ void k_prep_bn(float* alpha, float* beta, const float* g, const float* b,
                          const float* m, const float* v, const float* convb, int C) {
  int c = blockIdx.x * blockDim.x + threadIdx.x;
  if (c >= C) return;
  float sc = g[c] * rsqrtf(v[c] + EPSV);
  alpha[c] = sc;
  beta[c] = (convb ? convb[c] : 0.f) * sc + b[c] - m[c] * sc;
}

// depthwise 3x3, pad 1, fused affine + act
__global__ void k_dwconv3(float* __restrict__ out, const float* __restrict__ x,
                          const float* __restrict__ w, const float* __restrict__ alpha,
                          const float* __restrict__ beta, int act,
                          int B, int C, int H, int W) {
  size_t idx = (size_t)blockIdx.x * blockDim.x + threadIdx.x;
  size_t tot = (size_t)B * C * H * W;
  if (idx >= tot) return;
  int wv = (int)(idx % W);
  int hv = (int)((idx / W) % H);
  int c = (int)((idx / ((size_t)W * H)) % C);
  size_t base = (idx / ((size_t)W * H)) * (size_t)W * H;
  float s = 0.f;
#pragma unroll
  for (int ky = 0; ky < 3; ++ky) {
#pragma unroll
    for (int kx = 0; kx < 3; ++kx) {
      int iy = hv - 1 + ky, ix = wv - 1 + kx;
      if (iy >= 0 && iy < H && ix >= 0 && ix < W)
        s += w[c * 9 + ky * 3 + kx] * x[base + (size_t)iy * W + ix];
    }
  }
  float v = s * (alpha ? alpha[c] : 1.f) + (beta ? beta[c] : 0.f);
  out[idx] = actf(v, act);
}

__global__ void k_maxpool2(float* __restrict__ out, const float* __restrict__ x,
                           int B, int C, int H, int W) {
  int Ho = H / 2, Wo = W / 2;
  size_t idx = (size_t)blockIdx.x * blockDim.x + threadIdx.x;
  size_t tot = (size_t)B * C * Ho * Wo;
  if (idx >= tot) return;
  int ox = (int)(idx % Wo);
  int oy = (int)((idx / Wo) % Ho);
  size_t bc = idx / ((size_t)Wo * Ho);
  const float* p = x + bc * (size_t)H * W;
  int iy = oy * 2, ix = ox * 2;
  float m0 = fmaxf(p[(size_t)iy * W + ix], p[(size_t)iy * W + ix + 1]);
  float m1 = fmaxf(p[(size_t)(iy + 1) * W + ix], p[(size_t)(iy + 1) * W + ix + 1]);
  out[idx] = fmaxf(fmaxf(m0, m1), 0.f);
}

// bilinear 2x upsample (half-pixel), relu, optional residual skip
__global__ void k_up2(float* __restrict__ out, const float* __restrict__ x,
                      const float* __restrict__ skip, int B, int C, int H, int W) {
  int Ho = 2 * H, Wo = 2 * W;
  size_t idx = (size_t)blockIdx.x * blockDim.x + threadIdx.x;
  size_t tot = (size_t)B * C * Ho * Wo;
  if (idx >= tot) return;
  int ox = (int)(idx % Wo);
  int oy = (int)((idx / Wo) % Ho);
  size_t bc = idx / ((size_t)Wo * Ho);
  const float* p = x + bc * (size_t)H * W;
  float sy = oy * 0.5f - 0.25f;
  float sx = ox * 0.5f - 0.25f;
  int y0 = (int)floorf(sy), x0 = (int)floorf(sx);
  float fy = sy - y0, fx = sx - x0;
  int y0c = min(max(y0, 0), H - 1), y1c = min(max(y0 + 1, 0), H - 1);
  int x0c = min(max(x0, 0), W - 1), x1c = min(max(x0 + 1, 0), W - 1);
  float v00 = p[(size_t)y0c * W + x0c], v01 = p[(size_t)y0c * W + x1c];
  float v10 = p[(size_t)y1c * W + x0c], v11 = p[(size_t)y1c * W + x1c];
  float v = (1.f - fy) * ((1.f - fx) * v00 + fx * v01) + fy * ((1.f - fx) * v10 + fx * v11);
  v = fmaxf(v, 0.f);
  if (skip) v += skip[idx];
  out[idx] = v;
}

// layernorm over C.  Row r: base = (r/Nn)*C*Nn + r%Nn, channel stride = Nn.
__global__ void k_layernorm(float* __restrict__ out, const float* __restrict__ in,
                            const float* __restrict__ g, const float* __restrict__ b,
                            int rows, int C, int Nn) {
  int r = blockIdx.x;
  if (r >= rows) return;
  size_t base = (size_t)(r / Nn) * (size_t)C * Nn + (size_t)(r % Nn);
  __shared__ float red[128];
  int tid = threadIdx.x;
  float s = 0.f;
  for (int c = tid; c < C; c += 128) s += in[base + (size_t)c * Nn];
  red[tid] = s;
  __syncthreads();
  for (int o = 64; o > 0; o >>= 1) {
    if (tid < o) red[tid] += red[tid + o];
    __syncthreads();
  }
  float mu = red[0] / C;
  __syncthreads();
  s = 0.f;
  for (int c = tid; c < C; c += 128) {
    float d = in[base + (size_t)c * Nn] - mu;
    s += d * d;
  }
  red[tid] = s;
  __syncthreads();
  for (int o = 64; o > 0; o >>= 1) {
    if (tid < o) red[tid] += red[tid + o];
    __syncthreads();
  }
  float rstd = rsqrtf(red[0] / C + EPSV);
  for (int c = tid; c < C; c += 128) {
    float d = (in[base + (size_t)c * Nn] - mu) * rstd;
    out[base + (size_t)c * Nn] = d * g[c] + b[c];
  }
}

// NCHW -> [C][B*HW]
__global__ void k_tr_bc(float* __restrict__ out, const float* __restrict__ in,
                        int B, int C, int HW) {
  size_t idx = (size_t)blockIdx.x * blockDim.x + threadIdx.x;
  size_t tot = (size_t)B * C * HW;
  if (idx >= tot) return;
  size_t NB = (size_t)B * HW;
  int c = (int)(idx / NB);
  size_t rr = idx % NB;
  int b = (int)(rr / HW);
  int hw = (int)(rr % HW);
  out[idx] = in[((size_t)b * C + c) * HW + hw];
}

__global__ void k_add(float* __restrict__ out, const float* __restrict__ a,
                      const float* __restrict__ b, size_t n) {
  size_t idx = (size_t)blockIdx.x * blockDim.x + threadIdx.x;
  if (idx < n) out[idx] = a[idx] + b[idx];
}

// -------------------------- SS2D (selective scan) --------------------------
// Build 4 direction matrices: xs[k][c][b*L + l]
__global__ void k_build_dirs(float* __restrict__ xs, const float* __restrict__ x,
                             int B, int C, int H, int W) {
  int L = H * W;
  size_t BL = (size_t)B * L;
  size_t idx = (size_t)blockIdx.x * blockDim.x + threadIdx.x;
  size_t tot = (size_t)C * BL;
  if (idx >= tot) return;
  int c = (int)(idx / BL);
  size_t rr = idx % BL;
  int b = (int)(rr / L);
  int l = (int)(rr % L);
  int h = l / W, w = l % W;
  int lt = w * H + h;
  float v = x[((size_t)b * C + c) * L + l];
  size_t cb = (size_t)c * BL + (size_t)b * L;
  size_t stride = (size_t)C * BL;
  xs[0 * stride + cb + l] = v;
  xs[1 * stride + cb + lt] = v;
  xs[2 * stride + cb + (L - 1 - l)] = v;
  xs[3 * stride + cb + (L - 1 - lt)] = v;
}

__global__ void k_scan(const float* __restrict__ dts, const float* __restrict__ dtb,
                       const float* __restrict__ xs, const float* __restrict__ xdbl,
                       const float* __restrict__ alogs, const float* __restrict__ ds,
                       float* __restrict__ ys, int B, int K, int C, int L, int R, int Cx) {
  int id = blockIdx.x;  // b*(K*C) + k*C + c
  int c = id % C;
  int k = (id / C) % K;
  int b = id / (C * K);
  int lane = threadIdx.x;
  if (lane >= 16) return;
  size_t BL = (size_t)B * L;
  const float* dtr = dts + ((size_t)k * C + c) * BL + (size_t)b * L;
  const float* ur = xs + ((size_t)k * C + c) * BL + (size_t)b * L;
  const float* Br = xdbl + ((size_t)k * Cx + R + lane) * BL + (size_t)b * L;
  const float* Cr = xdbl + ((size_t)k * Cx + R + 16 + lane) * BL + (size_t)b * L;
  float* yr = ys + ((size_t)k * C + c) * BL + (size_t)b * L;
  float a = -__expf(alogs[((size_t)k * C + c) * 16 + lane]);
  float dtbias = dtb[k * C + c];
  float Dv = ds[k * C + c];
  float h = 0.f;
  for (int l = 0; l < L; ++l) {
    float d = softplusf(dtr[l] + dtbias);
    float u = ur[l];
    h = __expf(d * a) * h + (d * u) * Br[l];
    float p = h * Cr[l];
    p += __shfl_xor(p, 8, 16);
    p += __shfl_xor(p, 4, 16);
    p += __shfl_xor(p, 2, 16);
    p += __shfl_xor(p, 1, 16);
    if (lane == 0) yr[l] = p + Dv * u;
  }
}

__global__ void k_combine_dirs(float* __restrict__ out, const float* __restrict__ ys,
                               int B, int C, int H, int W) {
  int L = H * W;
  size_t BL = (size_t)B * L;
  size_t idx = (size_t)blockIdx.x * blockDim.x + threadIdx.x;
  size_t tot = (size_t)C * BL;
  if (idx >= tot) return;
  int c = (int)(idx / BL);
  size_t rr = idx % BL;
  int b = (int)(rr / L);
  int l = (int)(rr % L);
  int h = l / W, w = l % W;
  int lt = w * H + h;
  size_t cb = (size_t)c * BL + (size_t)b * L;
  size_t stride = (size_t)C * BL;
  float y = ys[0 * stride + cb + l] + ys[2 * stride + cb + (L - 1 - l)] +
            ys[1 * stride + cb + lt] + ys[3 * stride + cb + (L - 1 - lt)];
  out[idx] = y;
}

// y[c][b*L+l] *= silu(z_nchw[b][c][l])
__global__ void k_gate(float* __restrict__ y, const float* __restrict__ z,
                       int B, int C, int L) {
  size_t BL = (size_t)B * L;
  size_t idx = (size_t)blockIdx.x * blockDim.x + threadIdx.x;
  size_t tot = (size_t)C * BL;
  if (idx >= tot) return;
  int c = (int)(idx / BL);
  size_t rr = idx % BL;
  int b = (int)(rr / L);
  int l = (int)(rr % L);
  float zv = z[((size_t)b * C + c) * L + l];
  y[idx] *= zv * sigf(zv);
}

// ------------------------------- EMA kernels -------------------------------
__global__ void k_ema_means(float* __restrict__ hwcat, const float* __restrict__ x,
                            int H, int W) {
  int bgci = blockIdx.x;
  const float* p = x + (size_t)bgci * H * W;
  float* o = hwcat + (size_t)bgci * (H + W);
  int t = threadIdx.x;
  if (t < H) {
    float s = 0.f;
    for (int w = 0; w < W; ++w) s += p[(size_t)t * W + w];
    o[t] = s / W;
  }
  if (t < W) {
    float s = 0.f;
    for (int h = 0; h < H; ++h) s += p[(size_t)h * W + t];
    o[H + t] = s / H;
  }
}

__global__ void k_ema_conv1(float* __restrict__ out, const float* __restrict__ in,
                            const float* __restrict__ w, const float* __restrict__ b,
                            int bg, int cg, int P) {
  size_t idx = (size_t)blockIdx.x * blockDim.x + threadIdx.x;
  size_t tot = (size_t)bg * cg * P;
  if (idx >= tot) return;
  int p = (int)(idx % P);
  int co = (int)((idx / P) % cg);
  int bgi = (int)(idx / ((size_t)P * cg));
  float s = b[co];
  for (int ci = 0; ci < cg; ++ci)
    s += w[co * cg + ci] * in[((size_t)bgi * cg + ci) * P + p];
  out[idx] = s;
}

__global__ void k_ema_gate1(float* __restrict__ x1, const float* __restrict__ x,
                            const float* __restrict__ hwc, int H, int W, size_t tot) {
  size_t hw = (size_t)H * W;
  size_t idx = (size_t)blockIdx.x * blockDim.x + threadIdx.x;
  if (idx >= tot) return;
  size_t bgci = idx / hw;
  size_t i = idx % hw;
  int h = (int)(i / W), w = (int)(i % W);
  const float* o = hwc + bgci * (H + W);
  x1[idx] = x[idx] * sigf(o[h]) * sigf(o[H + w]);
}

__global__ void k_ema_gn(float* __restrict__ x1, const float* __restrict__ g,
                         const float* __restrict__ b, int cg, int hw) {
  int bgci = blockIdx.x;
  int ci = bgci % cg;
  float* p = x1 + (size_t)bgci * hw;
  __shared__ float red[256];
  int t = threadIdx.x;
  float s = 0.f;
  for (int i = t; i < hw; i += 256) s += p[i];
  red[t] = s;
  __syncthreads();
  for (int o = 128; o > 0; o >>= 1) {
    if (t < o) red[t] += red[t + o];
    __syncthreads();
  }
  float mu = red[0] / hw;
  __syncthreads();
  s = 0.f;
  for (int i = t; i < hw; i += 256) {
    float d = p[i] - mu;
    s += d * d;
  }
  red[t] = s;
  __syncthreads();
  for (int o = 128; o > 0; o >>= 1) {
    if (t < o) red[t] += red[t + o];
    __syncthreads();
  }
  float rstd = rsqrtf(red[0] / hw + EPSV);
  float gg = g[ci], bb = b[ci];
  for (int i = t; i < hw; i += 256) p[i] = (p[i] - mu) * rstd * gg + bb;
}

__global__ void k_ema_poolsm(float* __restrict__ sm, const float* __restrict__ x,
                             int cg, int hw) {
  int bgi = blockIdx.x;
  __shared__ float sums[32];
  int t = threadIdx.x;
  if (t < cg) {
    const float* p = x + ((size_t)bgi * cg + t) * hw;
    float s = 0.f;
    for (int i = 0; i < hw; ++i) s += p[i];
    sums[t] = s / hw;
  }
  __syncthreads();
  if (t == 0) {
    float mx = -1e30f;
    for (int c = 0; c < cg; ++c) mx = fmaxf(mx, sums[c]);
    float tot = 0.f;
    for (int c = 0; c < cg; ++c) {
      sums[c] = __expf(sums[c] - mx);
      tot += sums[c];
    }
    for (int c = 0; c < cg; ++c) sm[(size_t)bgi * cg + c] = sums[c] / tot;
  }
}

__global__ void k_ema_wts(float* __restrict__ out, const float* __restrict__ x,
                          const float* __restrict__ x11, const float* __restrict__ x21,
                          const float* __restrict__ x1, const float* __restrict__ x2,
                          int bg, int cg, int hw) {
  size_t idx = (size_t)blockIdx.x * blockDim.x + threadIdx.x;
  size_t tot = (size_t)bg * hw;
  if (idx >= tot) return;
  int bgi = (int)(idx / hw);
  int p = (int)(idx % hw);
  float s = 0.f;
  for (int c = 0; c < cg; ++c) {
    size_t off = ((size_t)bgi * cg + c) * hw + p;
    s += x11[(size_t)bgi * cg + c] * x2[off] + x21[(size_t)bgi * cg + c] * x1[off];
  }
  float gate = sigf(s);
  for (int c = 0; c < cg; ++c) {
    size_t off = ((size_t)bgi * cg + c) * hw + p;
    out[off] = x[off] * gate;
  }
}

// ------------------------------- KAN kernels -------------------------------
// Cubic B-spline bases over uniform knots t[j] = 0.4*j - 2.2 (j=0..11); 8 outputs.
__global__ void k_kan_basis(float* __restrict__ sXT, float* __restrict__ Bas,
                            const float* __restrict__ hin, int B, int C, int Nn) {
  size_t M = (size_t)B * Nn;
  size_t idx = (size_t)blockIdx.x * blockDim.x + threadIdx.x;
  size_t tot = (size_t)C * M;
  if (idx >= tot) return;
  int c = (int)(idx / M);
  size_t m = idx % M;
  int b = (int)(m / Nn);
  int n = (int)(m % Nn);
  float x = hin[((size_t)b * C + c) * Nn + n];
  sXT[(size_t)c * M + m] = x * sigf(x);
  float bas[11];
#pragma unroll
  for (int j = 0; j < 11; ++j) {
    float t0 = 0.4f * j - 2.2f;
    float t1 = t0 + 0.4f;
    bas[j] = (x >= t0 && x < t1) ? 1.f : 0.f;
  }
  for (int k = 1; k <= 3; ++k) {
    float inv = 1.f / (0.4f * k);
    for (int j = 0; j <= 10 - k; ++j) {
      float tj = 0.4f * j - 2.2f;
      float tjk1 = 0.4f * (j + k + 1) - 2.2f;
      bas[j] = (x - tj) * inv * bas[j] + (tjk1 - x) * inv * bas[j + 1];
    }
  }
#pragma unroll
  for (int j = 0; j < 8; ++j) Bas[((size_t)c * 8 + j) * M + m] = bas[j];
}

__global__ void k_kan_fold(float* __restrict__ sw, const float* __restrict__ spw,
                           const float* __restrict__ scl, int C) {
  size_t idx = (size_t)blockIdx.x * blockDim.x + threadIdx.x;
  size_t tot = (size_t)C * C * 8;
  if (idx >= tot) return;
  sw[idx] = spw[idx] * scl[idx / 8];
}

// ---------------------------------------------------------------------------
// Host-side orchestration
// ---------------------------------------------------------------------------
struct Pool {
  char* base;
  size_t off;
  float* f(size_t n) {
    size_t o = (off + 255) & ~(size_t)255;
    off = o + n * sizeof(float);
    return (float*)(base + o);
  }
};

static inline int cdiv(size_t a, int b) { return (int)((a + b - 1) / b); }

static void gemm(hipStream_t st, const float* A, const float* B, float* O,
                 int M, int N, int K, const float* alpha, const float* beta,
                 int act, int accum, int HW) {
  dim3 g((N + GT_N - 1) / GT_N, (M + GT_M - 1) / GT_M);
  k_gemm<<<g, 256, 0, st>>>(A, B, O, M, N, K, alpha, beta, act, accum, HW);
}

static void conv3x3(hipStream_t st, Pool sp, const float* x, float* out,
                    const float* w, const float* cb, const float* bng, const float* bnb,
                    const float* bnm, const float* bnv,
                    int B, int Ci, int H, int W, int Co, int stride, int act) {
  int Ho = (H - 1) / stride + 1, Wo = (W - 1) / stride + 1;
  const float* alpha = nullptr;
  const float* beta = cb;
  if (bng) {
    float* a = sp.f(Co);
    float* be = sp.f(Co);
    k_prep_bn<<<cdiv(Co, 256), 256, 0, st>>>(a, be, bng, bnb, bnm, bnv, cb, Co);
    alpha = a;
    beta = be;
  }
  size_t Ntot = (size_t)B * Ho * Wo;
  float* col = sp.f((size_t)Ci * 9 * Ntot);
  k_im2col<<<cdiv((size_t)Ci * 9 * Ntot, 256), 256, 0, st>>>(col, x, B, Ci, H, W, Ho, Wo, stride);
  gemm(st, w, col, out, Co, (int)Ntot, Ci * 9, alpha, beta, act, 0, Ho * Wo);
}

struct ConvLayerP {
  const float *c1w, *c1b, *b1g, *b1b, *b1m, *b1v;
  const float *c2w, *c2b, *b2g, *b2b, *b2m, *b2v;
};
struct PeP { const float *w, *b, *ng, *nb; };
struct KanLinP { const float *base_w, *spline_w, *scaler, *grid; };
struct DwP { const float *w, *b, *g, *bb, *m, *v; };
struct KanBlockP { const float *n2g, *n2b; KanLinP fc[3]; DwP dw[3]; };
struct SsP {
  const float *in_proj, *convw, *convb, *xproj, *dtw, *dtb, *alogs, *ds, *ong, *onb, *outproj;
};
struct EmaP { const float *c1w, *c1b, *c3w, *c3b, *gng, *gnb; };

static void conv_layer(hipStream_t st, Pool sp, const float* x, float* out,
                       const ConvLayerP& p, int B, int Ci, int Cm, int Co, int H, int W) {
  float* t = sp.f((size_t)B * Cm * H * W);
  conv3x3(st, sp, x, t, p.c1w, p.c1b, p.b1g, p.b1b, p.b1m, p.b1v, B, Ci, H, W, Cm, 1, ACT_RELU);
  conv3x3(st, sp, t, out, p.c2w, p.c2b, p.b2g, p.b2b, p.b2m, p.b2v, B, Cm, H, W, Co, 1, ACT_RELU);
}

static void ss2d(hipStream_t st, Pool sp, const float* x, float* out, const SsP& p,
                 int B, int dm, int H, int W) {
  int di = 2 * dm, L = H * W;
  int BL = B * L;
  int R = (dm + 15) / 16;
  if (R < 1) R = 1;
  int Cx = R + 32;
  float* xT = sp.f((size_t)dm * BL);
  k_tr_bc<<<cdiv((size_t)dm * BL, 256), 256, 0, st>>>(xT, x, B, dm, L);
  float* xc = sp.f((size_t)B * di * L);
  float* z = sp.f((size_t)B * di * L);
  gemm(st, p.in_proj, xT, xc, di, BL, dm, nullptr, nullptr, ACT_NONE, 0, L);
  gemm(st, p.in_proj + (size_t)di * dm, xT, z, di, BL, dm, nullptr, nullptr, ACT_NONE, 0, L);
  float* xn = sp.f((size_t)B * di * L);
  k_dwconv3<<<cdiv((size_t)B * di * L, 256), 256, 0, st>>>(xn, xc, p.convw, nullptr, p.convb,
                                                           ACT_SILU, B, di, H, W);
  float* xs = sp.f((size_t)4 * di * BL);
  k_build_dirs<<<cdiv((size_t)di * BL, 256), 256, 0, st>>>(xs, xn, B, di, H, W);
  float* xdbl = sp.f((size_t)4 * Cx * BL);
  for (int k = 0; k < 4; ++k)
    gemm(st, p.xproj + (size_t)k * Cx * di, xs + (size_t)k * di * BL,
         xdbl + (size_t)k * Cx * BL, Cx, BL, di, nullptr, nullptr, ACT_NONE, 0, BL);
  float* dts = sp.f((size_t)4 * di * BL);
  for (int k = 0; k < 4; ++k)
    gemm(st, p.dtw + (size_t)k * di * R, xdbl + (size_t)k * Cx * BL,
         dts + (size_t)k * di * BL, di, BL, R, nullptr, nullptr, ACT_NONE, 0, BL);
  float* ys = sp.f((size_t)4 * di * BL);
  k_scan<<<B * 4 * di, 32, 0, st>>>(dts, p.dtb, xs, xdbl, p.alogs, p.ds, ys, B, 4, di, L, R, Cx);
  float* ym = sp.f((size_t)di * BL);
  k_combine_dirs<<<cdiv((size_t)di * BL, 256), 256, 0, st>>>(ym, ys, B, di, H, W);
  k_layernorm<<<BL, 128, 0, st>>>(ym, ym, p.ong, p.onb, BL, di, BL);
  k_gate<<<cdiv((size_t)di * BL, 256), 256, 0, st>>>(ym, z, B, di, L);
  gemm(st, p.outproj, ym, out, dm, BL, di, nullptr, nullptr, ACT_NONE, 0, L);
}

static void ema(hipStream_t st, Pool sp, const float* x, float* out, const EmaP& p,
                int B, int C, int H, int W) {
  int cg = C / 8, bg = B * 8, hw = H * W;
  float* hwcat = sp.f((size_t)bg * cg * (H + W));
  k_ema_means<<<bg * cg, 128, 0, st>>>(hwcat, x, H, W);
  float* hwc = sp.f((size_t)bg * cg * (H + W));
  k_ema_conv1<<<cdiv((size_t)bg * cg * (H + W), 256), 256, 0, st>>>(hwc, hwcat, p.c1w, p.c1b,
                                                                     bg, cg, H + W);
  float* x1 = sp.f((size_t)bg * cg * hw);
  k_ema_gate1<<<cdiv((size_t)bg * cg * hw, 256), 256, 0, st>>>(x1, x, hwc, H, W,
                                                               (size_t)bg * cg * hw);
  k_ema_gn<<<bg * cg, 256, 0, st>>>(x1, p.gng, p.gnb, cg, hw);
  float* col = sp.f((size_t)cg * 9 * bg * hw);
  k_im2col<<<cdiv((size_t)cg * 9 * bg * hw, 256), 256, 0, st>>>(col, x, bg, cg, H, W, H, W, 1);
  float* x2 = sp.f((size_t)bg * cg * hw);
  gemm(st, p.c3w, col, x2, cg, bg * hw, cg * 9, nullptr, p.c3b, ACT_NONE, 0, hw);
  float* x11 = sp.f((size_t)bg * cg);
  float* x21 = sp.f((size_t)bg * cg);
  k_ema_poolsm<<<bg, 32, 0, st>>>(x11, x1, cg, hw);
  k_ema_poolsm<<<bg, 32, 0, st>>>(x21, x2, cg, hw);
  k_ema_wts<<<cdiv((size_t)bg * hw, 256), 256, 0, st>>>(out, x, x11, x21, x1, x2, bg, cg, hw);
}

static void attn(hipStream_t st, Pool sp, const float* x, float* out, const SsP& ss,
                 const EmaP& em, int B, int dm, int H, int W) {
  float* t = sp.f((size_t)B * dm * H * W);
  ss2d(st, sp, x, t, ss, B, dm, H, W);
  ema(st, sp, t, out, em, B, dm, H, W);
}

static void kan_block(hipStream_t st, Pool sp, float* tok, const KanBlockP& p,
                      int B, int C, int H, int W) {
  int Nn = H * W;
  size_t tokN = (size_t)B * C * Nn;
  size_t M = (size_t)B * Nn;
  float* hbuf = sp.f(tokN);
  float* h2 = sp.f(tokN);
  float* sXT = sp.f((size_t)C * M);
  float* Bas = sp.f((size_t)C * 8 * M);
  float* sw = sp.f((size_t)C * C * 8);
  float* al = sp.f(C);
  float* be = sp.f(C);
  k_layernorm<<<(int)M, 128, 0, st>>>(hbuf, tok, p.n2g, p.n2b, (int)M, C, Nn);
  for (int i = 0; i < 3; ++i) {
    k_kan_basis<<<cdiv((size_t)C * M, 256), 256, 0, st>>>(sXT, Bas, hbuf, B, C, Nn);
    k_kan_fold<<<cdiv((size_t)C * C * 8, 256), 256, 0, st>>>(sw, p.fc[i].spline_w,
                                                             p.fc[i].scaler, C);
    gemm(st, p.fc[i].base_w, sXT, h2, C, (int)M, C, nullptr, nullptr, ACT_NONE, 0, Nn);
    gemm(st, sw, Bas, h2, C, (int)M, C * 8, nullptr, nullptr, ACT_NONE, 1, Nn);
    k_prep_bn<<<cdiv(C, 256), 256, 0, st>>>(al, be, p.dw[i].g, p.dw[i].bb, p.dw[i].m,
                                            p.dw[i].v, p.dw[i].b, C);
    k_dwconv3<<<cdiv(tokN, 256), 256, 0, st>>>(hbuf, h2, p.dw[i].w, al, be, ACT_RELU, B, C, H, W);
  }
  k_add<<<cdiv(tokN, 256), 256, 0, st>>>(tok, tok, hbuf, tokN);
}

extern "C" void kernel_launch(void* const* d_in, const int* in_sizes, int n_in,
                              void* d_out, int out_size, void* d_ws, size_t ws_size,
                              hipStream_t stream) {
  (void)in_sizes; (void)n_in; (void)out_size; (void)ws_size;
  const float* const* IN = (const float* const*)d_in;
  int cur = 0;
  auto nx = [&]() { return IN[cur++]; };

  const float* X = nx();
  auto rdCL = [&](ConvLayerP& p) {
    p.c1w = nx(); p.c1b = nx(); p.b1g = nx(); p.b1b = nx(); p.b1m = nx(); p.b1v = nx();
    p.c2w = nx(); p.c2b = nx(); p.b2g = nx(); p.b2b = nx(); p.b2m = nx(); p.b2v = nx();
  };
  auto rdPE = [&](PeP& p) { p.w = nx(); p.b = nx(); p.ng = nx(); p.nb = nx(); };
  auto rdKB = [&](KanBlockP& p) {
    p.n2g = nx(); p.n2b = nx();
    for (int i = 0; i < 3; ++i) {
      p.fc[i].base_w = nx(); p.fc[i].spline_w = nx(); p.fc[i].scaler = nx(); p.fc[i].grid = nx();
    }
    for (int i = 0; i < 3; ++i) {
      p.dw[i].w = nx(); p.dw[i].b = nx();
      p.dw[i].g = nx(); p.dw[i].bb = nx(); p.dw[i].m = nx(); p.dw[i].v = nx();
    }
  };
  auto rdSS = [&](SsP& p) {
    p.in_proj = nx(); p.convw = nx(); p.convb = nx(); p.xproj = nx(); p.dtw = nx();
    p.dtb = nx(); p.alogs = nx(); p.ds = nx(); p.ong = nx(); p.onb = nx(); p.outproj = nx();
  };
  auto rdEMA = [&](EmaP& p) {
    p.c1w = nx(); p.c1b = nx(); p.c3w = nx(); p.c3b = nx(); p.gng = nx(); p.gnb = nx();
  };

  ConvLayerP enc1, enc2, enc3, dec1, dec2, dec3, dec4, dec5;
  PeP pe3, pe4;
  KanBlockP blk1, blk2;
  SsP ss1, ss2, ss3, ssd1, ssd2, ssd3, ssd4;
  EmaP em1, em2, em3, emd1, emd2, emd3, emd4;
  const float *n3g, *n3b, *n4g, *n4b, *finw, *finb;

  rdCL(enc1); rdCL(enc2); rdCL(enc3);
  rdPE(pe3); rdPE(pe4);
  rdKB(blk1); rdKB(blk2);
  n3g = nx(); n3b = nx(); n4g = nx(); n4b = nx();
  rdCL(dec1); rdCL(dec2); rdCL(dec3); rdCL(dec4); rdCL(dec5);
  finw = nx(); finb = nx();
  rdSS(ss1); rdSS(ss2); rdSS(ss3); rdSS(ssd1); rdSS(ssd2); rdSS(ssd3); rdSS(ssd4);
  rdEMA(em1); rdEMA(em2); rdEMA(em3); rdEMA(emd1); rdEMA(emd2); rdEMA(emd3); rdEMA(emd4);

  const int B = 2;
  Pool pool{(char*)d_ws, 0};
  // Persistent buffers
  float* o1 = pool.f((size_t)B * 16 * 128 * 128);
  float* t1 = pool.f((size_t)B * 16 * 128 * 128);
  float* o2 = pool.f((size_t)B * 32 * 64 * 64);
  float* t2 = pool.f((size_t)B * 32 * 64 * 64);
  float* o3 = pool.f((size_t)B * 128 * 32 * 32);
  float* t3 = pool.f((size_t)B * 128 * 32 * 32);
  float* t4 = pool.f((size_t)B * 160 * 16 * 16);
  float* bufA = pool.f((size_t)B * 16 * 256 * 256);
  float* bufB = pool.f((size_t)B * 16 * 256 * 256);
  hipStream_t st = stream;

  // ---- encoder ----
  {
    Pool sp = pool;
    float* t = sp.f((size_t)B * 16 * 256 * 256);
    conv_layer(st, sp, X, t, enc1, B, 3, 16, 16, 256, 256);
    k_maxpool2<<<cdiv((size_t)B * 16 * 128 * 128, 256), 256, 0, st>>>(o1, t, B, 16, 256, 256);
  }
  { Pool sp = pool; attn(st, sp, o1, t1, ss1, em1, B, 16, 128, 128); }
  {
    Pool sp = pool;
    float* t = sp.f((size_t)B * 32 * 128 * 128);
    conv_layer(st, sp, o1, t, enc2, B, 16, 32, 32, 128, 128);
    k_maxpool2<<<cdiv((size_t)B * 32 * 64 * 64, 256), 256, 0, st>>>(o2, t, B, 32, 128, 128);
  }
  { Pool sp = pool; attn(st, sp, o2, t2, ss2, em2, B, 32, 64, 64); }
  {
    Pool sp = pool;
    float* t = sp.f((size_t)B * 128 * 64 * 64);
    conv_layer(st, sp, o2, t, enc3, B, 32, 128, 128, 64, 64);
    k_maxpool2<<<cdiv((size_t)B * 128 * 32 * 32, 256), 256, 0, st>>>(o3, t, B, 128, 64, 64);
  }
  { Pool sp = pool; attn(st, sp, o3, t3, ss3, em3, B, 128, 32, 32); }

  // ---- bottleneck: patch embeds + KAN blocks ----
  {
    Pool sp = pool;
    conv3x3(st, sp, o3, t4, pe3.w, pe3.b, nullptr, nullptr, nullptr, nullptr,
            B, 128, 32, 32, 160, 2, ACT_NONE);
    k_layernorm<<<B * 256, 128, 0, st>>>(t4, t4, pe3.ng, pe3.nb, B * 256, 160, 256);
    kan_block(st, sp, t4, blk1, B, 160, 16, 16);
    k_layernorm<<<B * 256, 128, 0, st>>>(t4, t4, n3g, n3b, B * 256, 160, 256);
  }
  {
    Pool sp = pool;
    conv3x3(st, sp, t4, bufA, pe4.w, pe4.b, nullptr, nullptr, nullptr, nullptr,
            B, 160, 16, 16, 256, 2, ACT_NONE);
    k_layernorm<<<B * 64, 128, 0, st>>>(bufA, bufA, pe4.ng, pe4.nb, B * 64, 256, 64);
    kan_block(st, sp, bufA, blk2, B, 256, 8, 8);
    k_layernorm<<<B * 64, 128, 0, st>>>(bufA, bufA, n4g, n4b, B * 64, 256, 64);
  }

  // ---- decoder ----
  auto dec_stage = [&](const float* x, float* out, const ConvLayerP& p, int Ci, int Co,
                       int H, int W, const float* skip) {
    Pool sp = pool;
    float* t = sp.f((size_t)B * Co * H * W);
    conv_layer(st, sp, x, t, p, B, Ci, Ci, Co, H, W);
    k_up2<<<cdiv((size_t)B * Co * 4 * H * W, 256), 256, 0, st>>>(out, t, skip, B, Co, H, W);
  };

  dec_stage(bufA, bufB, dec1, 256, 160, 8, 8, t4);          // (B,160,16,16)
  { Pool sp = pool; attn(st, sp, bufB, bufA, ssd1, emd1, B, 160, 16, 16); }
  dec_stage(bufA, bufB, dec2, 160, 128, 16, 16, t3);        // (B,128,32,32)
  { Pool sp = pool; attn(st, sp, bufB, bufA, ssd2, emd2, B, 128, 32, 32); }
  dec_stage(bufA, bufB, dec3, 128, 32, 32, 32, t2);         // (B,32,64,64)
  { Pool sp = pool; attn(st, sp, bufB, bufA, ssd3, emd3, B, 32, 64, 64); }
  dec_stage(bufA, bufB, dec4, 32, 16, 64, 64, t1);          // (B,16,128,128)
  { Pool sp = pool; attn(st, sp, bufB, bufA, ssd4, emd4, B, 16, 128, 128); }
  dec_stage(bufA, bufB, dec5, 16, 16, 128, 128, nullptr);   // (B,16,256,256)

  // ---- final 1x1 conv to d_out ----
  {
    Pool sp = pool;
    int HW = 256 * 256;
    float* xT = sp.f((size_t)16 * B * HW);
    k_tr_bc<<<cdiv((size_t)16 * B * HW, 256), 256, 0, st>>>(xT, bufB, B, 16, HW);
    gemm(st, finw, xT, (float*)d_out, 9, B * HW, 16, nullptr, finb, ACT_NONE, 0, HW);
  }
}